// GFN1_3573412790701
// MI455X (gfx1250) — compile-verified
//
#include <hip/hip_runtime.h>
#include <cstdint>
#include <cmath>

// ---------------------------------------------------------------------------
// GFN1-style repulsion energy: per-edge gather + exp/sqrt + sorted segment sum
// Memory-bound (~77MB streamed -> ~3.3us floor @ 23.3 TB/s). Strategy:
//   * pre-collapse Z->alpha/Zeff indirection into an L2-resident float2 table
//   * stream edge tiles into LDS with CDNA5 async global->LDS b128 copies,
//     marked non-temporal (read-once data must not evict the gather table)
//   * exploit sorted idx_i: run-length accumulate, atomicAdd only at boundaries
// ---------------------------------------------------------------------------

constexpr int BLOCK = 256;           // 8 wave32 waves per block
constexpr int EPT   = 8;             // edges per thread (contiguous run)
constexpr int TILE  = BLOCK * EPT;   // 2048 edges per block -> 24KB LDS staging

__device__ __forceinline__ uint32_t lds_addr_of(const void* p) {
  // flat address of an LDS object: low 32 bits are the workgroup-relative
  // LDS byte offset (aperture encoding: {SHARED_BASE, offset[31:0]})
  return (uint32_t)(uintptr_t)p;
}

__device__ __forceinline__ void async_copy_b128_nt(uint32_t lds_byte_addr,
                                                   uint32_t global_byte_off,
                                                   const void* sbase) {
  // GVS mode: mem = SGPR base + VGPR 32-bit offset; 16B straight to LDS.
  // th:TH_LOAD_NT -> non-temporal: streamed once, keep L2 for the gather table.
  asm volatile("global_load_async_to_lds_b128 %0, %1, %2 th:TH_LOAD_NT"
               :: "v"(lds_byte_addr), "v"(global_byte_off), "s"(sbase)
               : "memory");
}

__device__ __forceinline__ void wait_async0() {
#if __has_builtin(__builtin_amdgcn_s_wait_asynccnt)
  __builtin_amdgcn_s_wait_asynccnt(0);
#else
  asm volatile("s_wait_asynccnt 0" ::: "memory");
#endif
}

// ---------------------------------------------------------------------------
__global__ __launch_bounds__(BLOCK)
void zero_f32(float* __restrict__ p, int n) {
  int i = blockIdx.x * BLOCK + threadIdx.x;
  if (i < n) p[i] = 0.0f;
}

// tab[a] = { sqrt(|alpha[Z[a]]|), |Zeff[Z[a]]| }  -- 800KB, L2-resident
__global__ __launch_bounds__(BLOCK)
void build_tab(const int* __restrict__ Z, const float* __restrict__ alpha,
               const float* __restrict__ Zeff, float2* __restrict__ tab,
               int nAtoms) {
  int a = blockIdx.x * BLOCK + threadIdx.x;
  if (a < nAtoms) {
    int z = Z[a];
    tab[a] = make_float2(__builtin_amdgcn_sqrtf(fabsf(alpha[z])),
                         fabsf(Zeff[z]));
  }
}

// ---------------------------------------------------------------------------
__global__ __launch_bounds__(BLOCK)
void edge_seg_kernel(const float*  __restrict__ Dij,
                     const int*    __restrict__ idxI,
                     const int*    __restrict__ idxJ,
                     const float2* __restrict__ tab,
                     float*        __restrict__ out,
                     int nEdges) {
  __shared__ alignas(16) float sD[TILE];
  __shared__ alignas(16) int   sI[TILE];
  __shared__ alignas(16) int   sJ[TILE];

  const int tid      = threadIdx.x;
  const int tileBase = blockIdx.x * TILE;

  if (tileBase + TILE <= nEdges) {
    // Full tile: CDNA5 async global->LDS DMA, 16B per lane-transfer.
    const void* gD = (const void*)(Dij  + tileBase);
    const void* gI = (const void*)(idxI + tileBase);
    const void* gJ = (const void*)(idxJ + tileBase);
    const uint32_t aD = lds_addr_of(sD);
    const uint32_t aI = lds_addr_of(sI);
    const uint32_t aJ = lds_addr_of(sJ);
#pragma unroll
    for (int it = 0; it < (TILE * 4) / (BLOCK * 16); ++it) {  // 2 iterations
      uint32_t off = (uint32_t)(tid + it * BLOCK) * 16u;
      async_copy_b128_nt(aD + off, off, gD);
      async_copy_b128_nt(aI + off, off, gI);
      async_copy_b128_nt(aJ + off, off, gJ);
    }
    wait_async0();   // per-wave ASYNCcnt drain; barrier below publishes LDS
  } else {
    // Tail tile (not hit for 6.4M edges, kept for robustness)
    for (int t = tid; t < TILE; t += BLOCK) {
      int e = tileBase + t;
      sD[t] = (e < nEdges) ? Dij[e]  : 1.0f;
      sI[t] = (e < nEdges) ? idxI[e] : -1;
      sJ[t] = (e < nEdges) ? idxJ[e] : 0;
    }
  }
  __syncthreads();

  // Each thread owns EPT consecutive (sorted-by-segment) edges.
  const int lbase = tid * EPT;
  int    curSeg = sI[lbase];
  float2 ti     = (curSeg >= 0) ? tab[curSeg] : make_float2(0.f, 0.f);
  float  acc    = 0.0f;

#pragma unroll
  for (int k = 0; k < EPT; ++k) {
    int ii = sI[lbase + k];
    if (ii < 0) break;                       // tail padding
    if (ii != curSeg) {                      // segment boundary: flush run
      atomicAdd(out + curSeg, acc);
      acc = 0.0f;
      curSeg = ii;
      ti = tab[ii];
    }
    float  d  = sD[lbase + k];
    float2 tj = tab[sJ[lbase + k]];
    // vij = |Zi||Zj|/d * exp(-sqrt(ai*aj*d^3)); sqrt factored via tab.x
    float sd = __builtin_amdgcn_sqrtf(d);
    float x  = (ti.x * tj.x) * (d * sd);
    float ex = __builtin_amdgcn_exp2f(x * -1.4426950408889634f);
    acc += (ti.y * tj.y) * __builtin_amdgcn_rcpf(d) * ex;
  }
  if (curSeg >= 0) atomicAdd(out + curSeg, acc);
}

// ---------------------------------------------------------------------------
// Fallback if workspace is too small for the atom table: keep per-Z tables in
// LDS and gather Z per endpoint (double indirection, still correct).
__global__ __launch_bounds__(BLOCK)
void edge_seg_fallback(const float* __restrict__ Dij,
                       const int*   __restrict__ idxI,
                       const int*   __restrict__ idxJ,
                       const int*   __restrict__ Z,
                       const float* __restrict__ alpha,
                       const float* __restrict__ Zeff,
                       float*       __restrict__ out,
                       int nEdges, int zmax) {
  __shared__ float sSA[128];
  __shared__ float sZE[128];
  __shared__ alignas(16) float sD[TILE];
  __shared__ alignas(16) int   sI[TILE];
  __shared__ alignas(16) int   sJ[TILE];

  const int tid      = threadIdx.x;
  const int tileBase = blockIdx.x * TILE;

  if (tid < zmax) {
    sSA[tid] = __builtin_amdgcn_sqrtf(fabsf(alpha[tid]));
    sZE[tid] = fabsf(Zeff[tid]);
  }
  for (int t = tid; t < TILE; t += BLOCK) {
    int e = tileBase + t;
    sD[t] = (e < nEdges) ? Dij[e]  : 1.0f;
    sI[t] = (e < nEdges) ? idxI[e] : -1;
    sJ[t] = (e < nEdges) ? idxJ[e] : 0;
  }
  __syncthreads();

  const int lbase = tid * EPT;
  int   curSeg = sI[lbase];
  float tix = 0.f, tiy = 0.f, acc = 0.f;
  if (curSeg >= 0) { int z = Z[curSeg]; tix = sSA[z]; tiy = sZE[z]; }

#pragma unroll
  for (int k = 0; k < EPT; ++k) {
    int ii = sI[lbase + k];
    if (ii < 0) break;
    if (ii != curSeg) {
      atomicAdd(out + curSeg, acc);
      acc = 0.0f; curSeg = ii;
      int z = Z[ii]; tix = sSA[z]; tiy = sZE[z];
    }
    float d  = sD[lbase + k];
    int   zj = Z[sJ[lbase + k]];
    float sd = __builtin_amdgcn_sqrtf(d);
    float x  = (tix * sSA[zj]) * (d * sd);
    float ex = __builtin_amdgcn_exp2f(x * -1.4426950408889634f);
    acc += (tiy * sZE[zj]) * __builtin_amdgcn_rcpf(d) * ex;
  }
  if (curSeg >= 0) atomicAdd(out + curSeg, acc);
}

// ---------------------------------------------------------------------------
extern "C" void kernel_launch(void* const* d_in, const int* in_sizes, int n_in,
                              void* d_out, int out_size, void* d_ws, size_t ws_size,
                              hipStream_t stream) {
  const int*   Z     = (const int*)  d_in[0];
  const float* Dij   = (const float*)d_in[1];
  const int*   idx_i = (const int*)  d_in[2];
  const int*   idx_j = (const int*)  d_in[3];
  const float* alpha = (const float*)d_in[4];
  const float* Zeff  = (const float*)d_in[5];

  const int nAtoms = in_sizes[0];
  const int nEdges = in_sizes[1];
  const int zmax   = in_sizes[4];
  float* out = (float*)d_out;

  // segment_sum semantics: atoms with no edges contribute 0 -> zero every call
  zero_f32<<<(out_size + BLOCK - 1) / BLOCK, BLOCK, 0, stream>>>(out, out_size);

  const int nBlocks = (nEdges + TILE - 1) / TILE;   // 3125 for 6.4M edges

  if (ws_size >= (size_t)nAtoms * sizeof(float2)) {
    float2* tab = (float2*)d_ws;
    build_tab<<<(nAtoms + BLOCK - 1) / BLOCK, BLOCK, 0, stream>>>(
        Z, alpha, Zeff, tab, nAtoms);
    edge_seg_kernel<<<nBlocks, BLOCK, 0, stream>>>(
        Dij, idx_i, idx_j, tab, out, nEdges);
  } else {
    edge_seg_fallback<<<nBlocks, BLOCK, 0, stream>>>(
        Dij, idx_i, idx_j, Z, alpha, Zeff, out, nEdges, zmax);
  }
}